// MC3DAD_ONNX_31250182045870
// MI455X (gfx1250) — compile-verified
//
#include <hip/hip_runtime.h>

// ---------------------------------------------------------------------------
// KNN (k=5) local-covariance trace, normalized by per-batch max.
// Distance tiles via V_WMMA_F32_16X16X4_F32 (K=3 padded to 4):
//   A row  (candidate) = ( x,  y,  z, |c|^2 )
//   B col  (query)     = (-2x,-2y,-2z, 1    )
//   C init             = |q_n|^2 broadcast per column
//   D[m][n] = |c_m|^2 - 2 c_m.q_n + |q_n|^2 = squared distance.
// Per-lane branchless top-5 with a per-tile min-reduction early-out.
// ---------------------------------------------------------------------------

typedef __attribute__((ext_vector_type(2))) float v2f;
typedef __attribute__((ext_vector_type(8))) float v8f;

#define B_BATCH 8
#define NPTS 4096
#define CHUNK 1024                 // candidate points staged in LDS per pass (16 KB)
#define NCHUNK (NPTS / CHUNK)
#define TILES_PER_CHUNK (CHUNK / 16)
#define THREADS 256                // 8 waves (wave32)
#define QPW 16                     // queries per wave (one WMMA column tile)
#define QPB ((THREADS / 32) * QPW) // 128 queries per block

__device__ __forceinline__ float bperm_f(int srcLane, float v) {
  return __int_as_float(__builtin_amdgcn_ds_bpermute(srcLane << 2, __float_as_int(v)));
}
__device__ __forceinline__ int bperm_i(int srcLane, int v) {
  return __builtin_amdgcn_ds_bpermute(srcLane << 2, v);
}

// Branchless sorted insertion into 5 scalar registers (bd0 <= ... <= bd4).
// Conditions are monotone (c0 => c1 => ... => c4); all read old state.
#define TOPK_INSERT(dv, iv)                                                   \
  do {                                                                        \
    const float _d = (dv); const int _i = (iv);                               \
    const bool c4 = _d < bd4, c3 = _d < bd3, c2 = _d < bd2,                   \
               c1 = _d < bd1, c0 = _d < bd0;                                  \
    bd4 = c3 ? bd3 : (c4 ? _d : bd4);  bi4 = c3 ? bi3 : (c4 ? _i : bi4);      \
    bd3 = c2 ? bd2 : (c3 ? _d : bd3);  bi3 = c2 ? bi2 : (c3 ? _i : bi3);      \
    bd2 = c1 ? bd1 : (c2 ? _d : bd2);  bi2 = c1 ? bi1 : (c2 ? _i : bi2);      \
    bd1 = c0 ? bd0 : (c1 ? _d : bd1);  bi1 = c0 ? bi0 : (c1 ? _i : bi1);      \
    bd0 = c0 ? _d : bd0;               bi0 = c0 ? _i : bi0;                   \
  } while (0)

__global__ void knn_init_kernel(int* __restrict__ maxbits) {
  if (threadIdx.x < B_BATCH) maxbits[threadIdx.x] = 0;
}

__global__ __launch_bounds__(THREADS)
void knn_trace_kernel(const float* __restrict__ pcd,
                      float* __restrict__ trace,
                      int* __restrict__ maxbits) {
  __shared__ float4 spts[CHUNK];   // (x, y, z, |p|^2) per candidate

  const int lane  = threadIdx.x & 31;
  const int wave  = threadIdx.x >> 5;
  const int batch = blockIdx.x >> 5;          // 32 blocks per batch
  const int qgrp  = blockIdx.x & 31;
  const int ncol  = lane & 15;                // WMMA column this lane owns
  const int hi    = (lane >= 16);
  const int q     = qgrp * QPB + wave * QPW + ncol;  // query idx within batch
  const int gq    = batch * NPTS + q;

  // Query-side operands (fixed for the whole scan).
  const float qx = pcd[gq * 3 + 0];
  const float qy = pcd[gq * 3 + 1];
  const float qz = pcd[gq * 3 + 2];
  const float cq = qx * qx + qy * qy + qz * qz;

  v2f bmat;                                    // B rows: (-2x,-2y | -2z, 1)
  bmat.x = hi ? (-2.0f * qz) : (-2.0f * qx);
  bmat.y = hi ? 1.0f         : (-2.0f * qy);
  v8f cmat = {cq, cq, cq, cq, cq, cq, cq, cq}; // C: |q_n|^2 broadcast per column

  float bd0 = 3.4e38f, bd1 = 3.4e38f, bd2 = 3.4e38f, bd3 = 3.4e38f, bd4 = 3.4e38f;
  int   bi0 = 0, bi1 = 0, bi2 = 0, bi3 = 0, bi4 = 0;

  // Per-lane LDS base for this lane's A-row fragment (low half: x,y; high: z,|c|^2).
  const float* sbase = (const float*)spts + (hi ? 2 : 0);

  for (int ch = 0; ch < NCHUNK; ++ch) {
    __syncthreads();
    // Stage CHUNK candidate points (+ squared norm) into LDS; prefetch next chunk.
    for (int t = threadIdx.x; t < CHUNK; t += THREADS) {
      const int gp = (batch * NPTS + ch * CHUNK + t) * 3;
      const float x = pcd[gp + 0];
      const float y = pcd[gp + 1];
      const float z = pcd[gp + 2];
      spts[t] = make_float4(x, y, z, x * x + y * y + z * z);
      if (ch + 1 < NCHUNK)
        __builtin_prefetch(&pcd[gp + CHUNK * 3], 0, 0);  // global_prefetch_b8
    }
    __syncthreads();

    // Software-pipelined tile loop: load A(t+1) while consuming WMMA(t).
    v2f a;
    {
      const float* sp = sbase + ncol * 4;
      a.x = sp[0]; a.y = sp[1];
    }
    for (int tile = 0; tile < TILES_PER_CHUNK; ++tile) {
      // D[m][n] = squared distance tile (full EXEC, uniform flow).
      v8f d = __builtin_amdgcn_wmma_f32_16x16x4_f32(
          false, a, false, bmat, (short)0, cmat, false, false);

      // Prefetch next tile's A fragment (wraps at end of chunk; harmless).
      const int nt = (tile + 1) & (TILES_PER_CHUNK - 1);
      const float* spn = sbase + (nt * 16 + ncol) * 4;
      v2f an; an.x = spn[0]; an.y = spn[1];

      // Early-out: can any of these 8 candidates enter the top-5?
      const float mall =
          fminf(fminf(fminf(d[0], d[1]), fminf(d[2], d[3])),
                fminf(fminf(d[4], d[5]), fminf(d[6], d[7])));
      if (mall < bd4) {
        const int base = ch * CHUNK + tile * 16 + (hi ? 8 : 0);
        TOPK_INSERT(d[0], base + 0);
        TOPK_INSERT(d[1], base + 1);
        TOPK_INSERT(d[2], base + 2);
        TOPK_INSERT(d[3], base + 3);
        TOPK_INSERT(d[4], base + 4);
        TOPK_INSERT(d[5], base + 5);
        TOPK_INSERT(d[6], base + 6);
        TOPK_INSERT(d[7], base + 7);
      }
      a = an;
    }
  }

  // Merge this lane's top-5 with partner lane (same query column, other half).
  const int ox = lane ^ 16;
  const float rd0 = bperm_f(ox, bd0), rd1 = bperm_f(ox, bd1), rd2 = bperm_f(ox, bd2),
              rd3 = bperm_f(ox, bd3), rd4 = bperm_f(ox, bd4);
  const int   ri0 = bperm_i(ox, bi0), ri1 = bperm_i(ox, bi1), ri2 = bperm_i(ox, bi2),
              ri3 = bperm_i(ox, bi3), ri4 = bperm_i(ox, bi4);
  TOPK_INSERT(rd0, ri0);
  TOPK_INSERT(rd1, ri1);
  TOPK_INSERT(rd2, ri2);
  TOPK_INSERT(rd3, ri3);
  TOPK_INSERT(rd4, ri4);

  if (lane < 16) {
    // trace = sum |p_i|^2 - |sum p_i|^2 / k   (k = 5)
    const float* pb = pcd + batch * NPTS * 3;
    const int ids[5] = {bi0, bi1, bi2, bi3, bi4};
    float sx = 0.f, sy = 0.f, sz = 0.f, ss = 0.f;
#pragma unroll
    for (int j = 0; j < 5; ++j) {
      const float* p = pb + ids[j] * 3;
      const float x = p[0], y = p[1], z = p[2];
      sx += x; sy += y; sz += z;
      ss += x * x + y * y + z * z;
    }
    const float tr = ss - (sx * sx + sy * sy + sz * sz) * 0.2f;
    trace[batch * NPTS + q] = tr;
    const float trc = tr < 0.f ? 0.f : tr;   // traces are >= 0 up to rounding
    atomicMax(&maxbits[batch], __float_as_int(trc));
  }
}

__global__ void knn_norm_kernel(const float* __restrict__ trace,
                                const int* __restrict__ maxbits,
                                float* __restrict__ out) {
  const int i = blockIdx.x * blockDim.x + threadIdx.x;
  if (i < B_BATCH * NPTS) {
    const float m = __int_as_float(maxbits[i >> 12]);
    out[i] = trace[i] / (m + 1e-8f);
  }
}

extern "C" void kernel_launch(void* const* d_in, const int* in_sizes, int n_in,
                              void* d_out, int out_size, void* d_ws, size_t ws_size,
                              hipStream_t stream) {
  const float* pcd = (const float*)d_in[0];     // [8, 4096, 3] f32; k==5 hardcoded
  float* trace   = (float*)d_ws;                               // 8*4096 f32
  int*   maxbits = (int*)((char*)d_ws + (size_t)B_BATCH * NPTS * sizeof(float));

  knn_init_kernel<<<1, 32, 0, stream>>>(maxbits);
  knn_trace_kernel<<<B_BATCH * NPTS / QPB, THREADS, 0, stream>>>(pcd, trace, maxbits);
  knn_norm_kernel<<<(B_BATCH * NPTS + 255) / 256, 256, 0, stream>>>(trace, maxbits,
                                                                    (float*)d_out);
}